// CompGCN_4398046511943
// MI455X (gfx1250) — compile-verified
//
#include <hip/hip_runtime.h>
#include <math.h>

typedef float v2f __attribute__((ext_vector_type(2)));
typedef float v8f __attribute__((ext_vector_type(8)));

#define NUM_ENT   100000
#define NUM_RELT  400      // 2 * NUM_REL
#define DIM       200
#define N_EDGES   640000
#define HALF_E    320000
#define BATCH     4096
#define NT        13       // ceil(200/16) column tiles
#define KSTEPS    50       // 200 / 4
#define WSTRIDE   208      // padded weight row stride (dwords), 208%64=16 -> conflict free
#define CSTRIDE   204      // padded comp row stride, 16 rows hit 16 distinct banks
#define TILES_PER_BLOCK 32 // 32 edge tiles * 16 edges = 512 edges per block

__device__ __forceinline__ v8f wmma4(v2f a, v2f b, v8f c) {
    // V_WMMA_F32_16X16X4_F32: D = A(16x4) * B(4x16) + C, full f32 precision
    return __builtin_amdgcn_wmma_f32_16x16x4_f32(
        /*neg_a=*/false, a, /*neg_b=*/false, b,
        /*c_mod=*/(short)0, c, /*reuse_a=*/false, /*reuse_b=*/false);
}

// ---------------------------------------------------------------------------
// agg[v] = ((x[v] * loop_rel) @ w_loop) / 3 + bias   (initializes accumulator)
// One wave per 16x16 output tile. grid = (6250, 13), block = 32.
// B column index is CLAMPED (not masked) so the K-loop has no exec-mask
// branches; garbage columns >= DIM are dropped at the store.
// ---------------------------------------------------------------------------
__global__ __launch_bounds__(32)
void loop_init_kernel(const float* __restrict__ x, const float* __restrict__ w_loop,
                      const float* __restrict__ loop_rel, const float* __restrict__ bias,
                      float* __restrict__ agg) {
    const int lane = threadIdx.x;
    const int r0   = blockIdx.x * 16;
    const int n0   = blockIdx.y * 16;
    const int hi   = lane >> 4;
    const int ln   = lane & 15;
    const int c0   = 2 * hi;
    const int arow = r0 + ln;
    const int bn   = n0 + ln;
    const int bnc  = bn < DIM ? bn : (DIM - 1);   // clamp -> unconditional loads

    v8f acc = {};
    for (int ks = 0; ks < KSTEPS; ++ks) {
        const int k0 = ks * 4;
        v2f a = *(const v2f*)&x[(long)arow * DIM + k0 + c0];
        v2f l = *(const v2f*)&loop_rel[k0 + c0];
        a *= l;
        v2f b;
        b.x = w_loop[(k0 + c0) * DIM + bnc];
        b.y = w_loop[(k0 + c0 + 1) * DIM + bnc];
        acc = wmma4(a, b, acc);
    }
    if (bn < DIM) {
#pragma unroll
        for (int i = 0; i < 8; ++i) {
            const int row = r0 + i + 8 * hi;     // C layout: elem i -> M = i + 8*half
            agg[(long)row * DIM + bn] = acc[i] * (1.0f / 3.0f) + bias[bn];
        }
    }
}

// ---------------------------------------------------------------------------
// r_out = r_in @ w_rel.  grid = (25, 13), block = 32 (one wave / 16x16 tile).
// ---------------------------------------------------------------------------
__global__ __launch_bounds__(32)
void rel_gemm_kernel(const float* __restrict__ r_in, const float* __restrict__ w_rel,
                     float* __restrict__ r_out) {
    const int lane = threadIdx.x;
    const int r0   = blockIdx.x * 16;
    const int n0   = blockIdx.y * 16;
    const int hi   = lane >> 4;
    const int ln   = lane & 15;
    const int c0   = 2 * hi;
    const int arow = r0 + ln;
    const int bn   = n0 + ln;
    const int bnc  = bn < DIM ? bn : (DIM - 1);

    v8f acc = {};
    for (int ks = 0; ks < KSTEPS; ++ks) {
        const int k0 = ks * 4;
        v2f a = *(const v2f*)&r_in[(long)arow * DIM + k0 + c0];
        v2f b;
        b.x = w_rel[(k0 + c0) * DIM + bnc];
        b.y = w_rel[(k0 + c0 + 1) * DIM + bnc];
        acc = wmma4(a, b, acc);
    }
    if (bn < DIM) {
#pragma unroll
        for (int i = 0; i < 8; ++i)
            r_out[(long)(r0 + i + 8 * hi) * DIM + bn] = acc[i];
    }
}

// ---------------------------------------------------------------------------
// Edge message kernel. Per block: one 200x200 weight cached in LDS (166 KB),
// 8 waves x 4 edge tiles of 16 edges. comp = x[src]*r[etype] staged per-wave
// in LDS; N-tiles processed in PAIRS (A fragment reused for 2 WMMAs);
// dst/enorm hoisted into registers per edge tile; scatter via atomic f32 add.
// ---------------------------------------------------------------------------
__global__ __launch_bounds__(256)
void edge_msg_kernel(const float* __restrict__ x, const float* __restrict__ r,
                     const int* __restrict__ src, const int* __restrict__ dst,
                     const int* __restrict__ etype, const float* __restrict__ enorm,
                     const float* __restrict__ w_in, const float* __restrict__ w_out,
                     float* __restrict__ agg) {
    __shared__ float w_lds[DIM * WSTRIDE];          // 166,400 B
    __shared__ float comp_lds[8][16 * CSTRIDE];     // 104,448 B  (total 270,848 < 320 KB)

    const int tid   = threadIdx.x;
    const int lane  = tid & 31;
    const int wave  = tid >> 5;
    const int ebase = blockIdx.x * (TILES_PER_BLOCK * 16);
    // 320000 % 512 == 0 -> every block lies entirely in one direction half
    const float* __restrict__ w = (ebase < HALF_E) ? w_in : w_out;

    // Cooperative load of the 200x200 weight into LDS (zero the pad columns).
    for (int i = tid; i < DIM * WSTRIDE; i += 256) {
        const int row = i / WSTRIDE;
        const int col = i - row * WSTRIDE;
        w_lds[i] = (col < DIM) ? w[row * DIM + col] : 0.0f;
    }
    __syncthreads();

    const int hi = lane >> 4;
    const int ln = lane & 15;
    const int c0 = 2 * hi;
    float* cslab = comp_lds[wave];
    const float* arow = &cslab[ln * CSTRIDE + c0];   // this lane's A stream base

    for (int t = 0; t < 4; ++t) {
        const int e0 = ebase + (wave * 4 + t) * 16;

        // Gather composition rows (float2-vectorized) into this wave's slab.
        for (int i = lane; i < 16 * (DIM / 2); i += 32) {
            const int  rr = i / (DIM / 2);
            const int  kk = (i - rr * (DIM / 2)) * 2;
            const int  e  = e0 + rr;
            const long sx = (long)src[e]   * DIM + kk;
            const long sr = (long)etype[e] * DIM + kk;
            v2f cv = (*(const v2f*)&x[sx]) * (*(const v2f*)&r[sr]);
            *(v2f*)&cslab[rr * CSTRIDE + kk] = cv;
        }
        __syncthreads();   // uniform across all 8 waves (each runs 4 tiles)

        // Hoist per-edge dst/enorm for this lane's 8 output rows.
        int   dvec[8];
        float evec[8];
#pragma unroll
        for (int i = 0; i < 8; ++i) {
            const int e = e0 + i + 8 * hi;
            dvec[i] = dst[e];
            evec[i] = enorm[e];
        }

        // ---- paired N-tiles: 6 pairs covering tiles 0..11 ----
        for (int np = 0; np < 6; ++np) {
            const int bn0 = np * 32 + ln;        // tile 2*np
            const int bn1 = bn0 + 16;            // tile 2*np+1
            v8f acc0 = {}, acc1 = {};
#pragma unroll 2
            for (int ks = 0; ks < KSTEPS; ++ks) {
                const int k0 = ks * 4;
                v2f a = *(const v2f*)&arow[k0];   // one A fragment, two WMMAs
                v2f b0, b1;
                b0.x = w_lds[(k0 + c0) * WSTRIDE + bn0];
                b0.y = w_lds[(k0 + c0 + 1) * WSTRIDE + bn0];
                b1.x = w_lds[(k0 + c0) * WSTRIDE + bn1];
                b1.y = w_lds[(k0 + c0 + 1) * WSTRIDE + bn1];
                acc0 = wmma4(a, b0, acc0);
                acc1 = wmma4(a, b1, acc1);
            }
#pragma unroll
            for (int i = 0; i < 8; ++i) {
                atomicAdd(&agg[(long)dvec[i] * DIM + bn0], acc0[i] * evec[i]);
                atomicAdd(&agg[(long)dvec[i] * DIM + bn1], acc1[i] * evec[i]);
            }
        }

        // ---- remainder tile 12 (cols 192..207, only ln<8 valid) ----
        {
            const int bn = 192 + ln;             // <= 207: inside padded LDS (pad = 0)
            v8f acc = {};
#pragma unroll 2
            for (int ks = 0; ks < KSTEPS; ++ks) {
                const int k0 = ks * 4;
                v2f a = *(const v2f*)&arow[k0];
                v2f b;
                b.x = w_lds[(k0 + c0) * WSTRIDE + bn];
                b.y = w_lds[(k0 + c0 + 1) * WSTRIDE + bn];
                acc = wmma4(a, b, acc);
            }
            if (bn < DIM) {
#pragma unroll
                for (int i = 0; i < 8; ++i)
                    atomicAdd(&agg[(long)dvec[i] * DIM + bn], acc[i] * evec[i]);
            }
        }
        __syncthreads();
    }
}

// ---------------------------------------------------------------------------
// h = tanh(agg * 1/sqrt(1+eps))  (in place)
// ---------------------------------------------------------------------------
__global__ __launch_bounds__(256)
void finish_kernel(float* __restrict__ agg, long n) {
    const long i = (long)blockIdx.x * 256 + threadIdx.x;
    if (i < n) agg[i] = tanhf(agg[i] * 0.9999950000374997f);
}

// ---------------------------------------------------------------------------
// out = concat(x2[subj], r2[rel], x2[obj])
// ---------------------------------------------------------------------------
__global__ __launch_bounds__(256)
void gather_out_kernel(const float* __restrict__ x2, const float* __restrict__ r2,
                       const int* __restrict__ subj, const int* __restrict__ rel,
                       const int* __restrict__ obj, float* __restrict__ out) {
    const int i = blockIdx.x * 256 + threadIdx.x;
    if (i >= 3 * BATCH * DIM) return;
    const int sec = i / (BATCH * DIM);
    const int j   = i - sec * (BATCH * DIM);
    const int b   = j / DIM;
    const int d   = j - b * DIM;
    const int* idx = (sec == 0) ? subj : (sec == 1 ? rel : obj);
    const float* buf = (sec == 1) ? r2 : x2;
    out[i] = buf[(long)idx[b] * DIM + d];
}

extern "C" void kernel_launch(void* const* d_in, const int* in_sizes, int n_in,
                              void* d_out, int out_size, void* d_ws, size_t ws_size,
                              hipStream_t stream) {
    const int*   src       = (const int*)  d_in[0];
    const int*   dst       = (const int*)  d_in[1];
    const int*   etype     = (const int*)  d_in[2];
    const float* enorm     = (const float*)d_in[3];
    const int*   subj      = (const int*)  d_in[4];
    const int*   rel       = (const int*)  d_in[5];
    const int*   obj       = (const int*)  d_in[6];
    const float* init_emb  = (const float*)d_in[7];
    const float* init_rel  = (const float*)d_in[8];
    const float* w_in1     = (const float*)d_in[9];
    const float* w_out1    = (const float*)d_in[10];
    const float* w_loop1   = (const float*)d_in[11];
    const float* w_rel1    = (const float*)d_in[12];
    const float* loop_rel1 = (const float*)d_in[13];
    const float* bias1     = (const float*)d_in[14];
    const float* w_in2     = (const float*)d_in[15];
    const float* w_out2    = (const float*)d_in[16];
    const float* w_loop2   = (const float*)d_in[17];
    const float* w_rel2    = (const float*)d_in[18];
    const float* loop_rel2 = (const float*)d_in[19];
    const float* bias2     = (const float*)d_in[20];

    float* ws = (float*)d_ws;
    float* x1 = ws;                                      // 20,000,000 floats
    float* x2 = ws + 20000000L;                          // 20,000,000 floats
    float* r1 = ws + 40000000L;                          //     80,000 floats
    float* r2 = ws + 40080000L;                          //     80,000 floats
    float* out = (float*)d_out;

    const dim3 gTile(NUM_ENT / 16, NT);                  // 6250 x 13
    const dim3 gRel(NUM_RELT / 16, NT);                  //   25 x 13
    const int  edgeBlocks   = N_EDGES / (TILES_PER_BLOCK * 16);   // 1250
    const int  finishBlocks = (int)(((long)NUM_ENT * DIM + 255) / 256);
    const int  gatherBlocks = (3 * BATCH * DIM + 255) / 256;

    // ---- Layer 1 ----
    loop_init_kernel<<<gTile, 32, 0, stream>>>(init_emb, w_loop1, loop_rel1, bias1, x1);
    edge_msg_kernel<<<edgeBlocks, 256, 0, stream>>>(init_emb, init_rel, src, dst, etype,
                                                    enorm, w_in1, w_out1, x1);
    finish_kernel<<<finishBlocks, 256, 0, stream>>>(x1, (long)NUM_ENT * DIM);
    rel_gemm_kernel<<<gRel, 32, 0, stream>>>(init_rel, w_rel1, r1);

    // ---- Layer 2 ----
    loop_init_kernel<<<gTile, 32, 0, stream>>>(x1, w_loop2, loop_rel2, bias2, x2);
    edge_msg_kernel<<<edgeBlocks, 256, 0, stream>>>(x1, r1, src, dst, etype,
                                                    enorm, w_in2, w_out2, x2);
    finish_kernel<<<finishBlocks, 256, 0, stream>>>(x2, (long)NUM_ENT * DIM);
    rel_gemm_kernel<<<gRel, 32, 0, stream>>>(r1, w_rel2, r2);

    // ---- Output gathers ----
    gather_out_kernel<<<gatherBlocks, 256, 0, stream>>>(x2, r2, subj, rel, obj, out);
}